// pH_SHND_Core_5059471475348
// MI455X (gfx1250) — compile-verified
//
#include <hip/hip_runtime.h>
#include <math.h>

// ---------------- problem constants ----------------
#define MU_C  0.1f
#define EPS_C 0.01f

typedef __attribute__((ext_vector_type(2))) float v2f;
typedef __attribute__((ext_vector_type(8))) float v8f;

// ---------------- LDS layout (float offsets) ----------------
enum : int {
  OWH1  = 0,              // Wh1  [128][16]
  OWH2  = OWH1  + 2048,   // Wh2  [16][128]
  OBH1  = OWH2  + 2048,   // bh1  [128]
  OW1JR = OBH1  + 128,    // W1_jr[64][16]
  OB1JR = OW1JR + 1024,   // b1_jr[64]
  OW2JR = OB1JR + 64,     // W2_jr[64][64]
  OB2JR = OW2JR + 4096,   // b2_jr[64]
  OB3JR = OB2JR + 64,     // b3_jr[512]
  OW1B  = OB3JR + 512,    // W1_b [64][16]
  OB1B  = OW1B  + 1024,   // b1_b [64]
  OW2B  = OB1B  + 64,     // W2_b [64][64]
  OB2B  = OW2B  + 4096,   // b2_b [64]
  OB3B  = OB2B  + 64,     // b3_b [128]
  WTOT  = OB3B  + 128     // = 15360 floats
};
enum : int {
  ZT  = 0,          // 16x16   current z (z, then z1)
  TB  = 256,        // 16x128  tanh buf (reused as h1, ld=64)
  VB  = TB + 2048,  // 16x128  v buf    (reused as h2, ld=64)
  FB  = VB + 2048,  // 16x16   f
  GH  = FB + 256,   // 16x16   gH
  CH  = GH + 256,   // 16x16   Jm-chunk staging
  RF  = CH + 256,   // 16x16x16 Rf[b][r][k]
  WBF = RF + 4096,  // 16x16   w = Rf^T gH
  BO  = WBF + 256,  // 16x128  B-MLP out (Bmat flat)
  UB  = BO + 2048,  // 16x8    u tile
  WAVE_FLOATS = UB + 128    // = 11648 floats
};

// ---------------- fast tanh (gfx1250 V_TANH_F32) ----------------
__device__ __forceinline__ float fast_tanh(float x) {
#if __has_builtin(__builtin_amdgcn_tanhf)
  return __builtin_amdgcn_tanhf(x);
#elif __has_builtin(__builtin_amdgcn_tanh_f32)
  return __builtin_amdgcn_tanh_f32(x);
#else
  float e = __builtin_amdgcn_exp2f(x * 2.8853900817779268f);  // exp(2x)
  return 1.f - 2.f * __builtin_amdgcn_rcpf(e + 1.f);
#endif
}

// ---------------- Tensor Data Mover weight staging ----------------
#if __has_builtin(__builtin_amdgcn_tensor_load_to_lds)
#define HAVE_TDM 1
typedef __attribute__((ext_vector_type(4))) unsigned int u32x4;
typedef __attribute__((ext_vector_type(8))) int i32x8;
typedef __attribute__((ext_vector_type(4))) int i32x4;

// One 1D D# per array: tensor_dim0 = tile_dim0 = nfloats (<=65535), dim1 = 1,
// data_size = 4B, no multicast/pad/iterate.  Groups 2/3 zero (<=2D tensor).
// This toolchain's builtin is the 6-arg form:
//   (u32x4 g0, i32x8 g1, i32x4 g2, i32x4 g3, i32x8 pad, i32 cpol)
__device__ __forceinline__ void tdm_copy_to_lds(const float* gsrc,
                                                unsigned ldsByteOff,
                                                unsigned nfloats) {
  unsigned long long ga = (unsigned long long)(__SIZE_TYPE__)gsrc;
  u32x4 g0;
  g0[0] = 1u;                                                  // count=1, user D#
  g0[1] = ldsByteOff;                                          // lds_addr (bytes)
  g0[2] = (unsigned)ga;                                        // global_addr[31:0]
  g0[3] = (unsigned)((ga >> 32) & 0x1FFFFFFull) | (2u << 30);  // ga[56:32] | type=2
  i32x8 g1;
  g1[0] = 0x20000;                                    // data_size=2 (4B), mask=0
  g1[1] = (int)((nfloats & 0xFFFFu) << 16);           // tensor_dim0[15:0]
  g1[2] = (int)(((nfloats >> 16) & 0xFFFFu) | (1u << 16)); // td0[31:16] | td1=1
  g1[3] = (int)((nfloats & 0xFFFFu) << 16);           // td1 hi=0 | tile_dim0
  g1[4] = 1;                                          // tile_dim1=1, tile_dim2=0
  g1[5] = (int)nfloats;                               // tensor_dim0_stride lo
  g1[6] = 0;                                          // stride0 hi | stride1 lo
  g1[7] = 0;                                          // stride1 hi
  i32x4 gz4 = {0, 0, 0, 0};
  i32x8 gz8 = {0, 0, 0, 0, 0, 0, 0, 0};
  __builtin_amdgcn_tensor_load_to_lds(g0, g1, gz4, gz4, gz8, 0);
}
#else
#define HAVE_TDM 0
#endif

// ---------------- WMMA helpers (fp32, 16x16x4) ----------------
__device__ __forceinline__ v8f wmma4(v2f a, v2f b, v8f c) {
  return __builtin_amdgcn_wmma_f32_16x16x4_f32(false, a, false, b, (short)0, c,
                                               false, false);
}

// A fragment 16x4 fp32: lanes 0-15 M=lane (K=k0,k0+1), lanes 16-31 M=lane-16 (K=k0+2,k0+3)
__device__ __forceinline__ v2f lda_frag(const float* T, int ld, int lane, int k0) {
  int m  = lane & 15;
  int ko = k0 + ((lane >> 4) << 1);
  const float* p = T + m * ld + ko;
  v2f a; a.x = p[0]; a.y = p[1]; return a;
}
// B fragment 4x16 from W stored [N][K] used as B[k][n]=W[n][k]  (x @ W^T)
__device__ __forceinline__ v2f ldb_T(const float* W, int ldk, int lane, int n0, int k0) {
  int n  = n0 + (lane & 15);
  int ko = k0 + ((lane >> 4) << 1);
  const float* p = W + n * ldk + ko;
  v2f b; b.x = p[0]; b.y = p[1]; return b;
}
// B fragment 4x16 from W stored [K][N] used directly: B[k][n]=W[k][n]
__device__ __forceinline__ v2f ldb_N(const float* W, int ldn, int lane, int n0, int k0) {
  int n  = n0 + (lane & 15);
  int ko = k0 + ((lane >> 4) << 1);
  v2f b; b.x = W[ko * ldn + n]; b.y = W[(ko + 1) * ldn + n]; return b;
}

template <int K, bool BT>
__device__ __forceinline__ v8f gemm_tile(const float* A, int lda, const float* W,
                                         int ldw, int lane, int n0) {
  v8f acc = {};
#pragma unroll 8
  for (int k = 0; k < K; k += 4) {
    v2f a = lda_frag(A, lda, lane, k);
    v2f b = BT ? ldb_T(W, ldw, lane, n0, k) : ldb_N(W, ldw, lane, n0, k);
    acc = wmma4(a, b, acc);
  }
  return acc;
}

// D tile store: lane owns column n=lane&15, rows M=(lane>>4)*8 + v
__device__ __forceinline__ void st_tanh_bias(float* T, int ld, int lane, v8f d,
                                             const float* bias, int n0) {
  int n = lane & 15, moff = (lane >> 4) * 8;
  float bn = bias[n0 + n];
#pragma unroll
  for (int v = 0; v < 8; v++) T[(moff + v) * ld + n0 + n] = fast_tanh(d[v] + bn);
}

// ---------------- one RHS evaluation (inlined for addrspace inference) -------
__device__ __forceinline__
v8f eval_step(float* __restrict__ sw, float* __restrict__ wl,
              const float* __restrict__ w3jr, const float* __restrict__ w3b,
              int lane) {
  float* zt = wl + ZT;  float* tb = wl + TB;  float* vb = wl + VB;
  float* fb = wl + FB;  float* gH = wl + GH;  float* ch = wl + CH;
  float* rf = wl + RF;  float* wbf = wl + WBF; float* bo = wl + BO;
  float* ub = wl + UB;

  const int n = lane & 15, moff = (lane >> 4) * 8;
  const int b = lane & 15, i0 = (lane >> 4) * 8;

  // ---- Hamiltonian gradient: gH = MU*f + ((1-t^2) .* (f@Wh2)) @ Wh1 ----
#pragma unroll 2
  for (int nt = 0; nt < 8; nt++) {     // t = tanh(z @ Wh1^T + bh1)   [16,128]
    v8f d = gemm_tile<16, true>(zt, 16, sw + OWH1, 16, lane, nt * 16);
    st_tanh_bias(tb, 128, lane, d, sw + OBH1, nt * 16);
  }
  {                                    // f = MU*z + t @ Wh2^T        [16,16]
    v8f d = gemm_tile<128, true>(tb, 128, sw + OWH2, 128, lane, 0);
#pragma unroll
    for (int v = 0; v < 8; v++)
      fb[(moff + v) * 16 + n] = d[v] + MU_C * zt[(moff + v) * 16 + n];
  }
#pragma unroll 2
  for (int nt = 0; nt < 8; nt++) {     // vv = (1-t^2) .* (f @ Wh2)   [16,128]
    v8f d = gemm_tile<16, false>(fb, 16, sw + OWH2, 128, lane, nt * 16);
#pragma unroll
    for (int v = 0; v < 8; v++) {
      float t = tb[(moff + v) * 128 + nt * 16 + n];
      vb[(moff + v) * 128 + nt * 16 + n] = d[v] * (1.f - t * t);
    }
  }
  {                                    // gH = MU*f + vv @ Wh1        [16,16]
    v8f d = gemm_tile<128, false>(vb, 128, sw + OWH1, 16, lane, 0);
#pragma unroll
    for (int v = 0; v < 8; v++)
      gH[(moff + v) * 16 + n] = d[v] + MU_C * fb[(moff + v) * 16 + n];
  }

  // ---- JR MLP hidden layers (h1 -> tb ld64, h2 -> vb ld64) ----
#pragma unroll 2
  for (int nt = 0; nt < 4; nt++) {
    v8f d = gemm_tile<16, true>(zt, 16, sw + OW1JR, 16, lane, nt * 16);
    st_tanh_bias(tb, 64, lane, d, sw + OB1JR, nt * 16);
  }
#pragma unroll 2
  for (int nt = 0; nt < 4; nt++) {
    v8f d = gemm_tile<64, true>(tb, 64, sw + OW2JR, 64, lane, nt * 16);
    st_tanh_bias(vb, 64, lane, d, sw + OB2JR, nt * 16);
  }

  // ---- stream JR layer 3: 32 chunks of 16 cols; fold J and R on the fly ----
  float dzr[8];
  float wacc[8];
#pragma unroll
  for (int v = 0; v < 8; v++) { dzr[v] = 0.f; wacc[v] = 0.f; }

#pragma unroll 1
  for (int c = 0; c < 32; c++) {
    v8f d = gemm_tile<64, true>(vb, 64, w3jr, 64, lane, c * 16);
    float bn = sw[OB3JR + c * 16 + n];
    if (c < 16) {
      // chunk = Jm[:, c, :]
#pragma unroll
      for (int v = 0; v < 8; v++) ch[(moff + v) * 16 + n] = d[v] + bn;
      float s = 0.f;
#pragma unroll
      for (int j = 0; j < 16; j++) s += ch[b * 16 + j] * gH[b * 16 + j];
#pragma unroll
      for (int v = 0; v < 8; v++) if (c == i0 + v) dzr[v] += s;
      float ghc = gH[b * 16 + c];
#pragma unroll
      for (int v = 0; v < 8; v++) dzr[v] -= ch[b * 16 + i0 + v] * ghc;
    } else {
      int r = c - 16;  // chunk = Rf[:, r, :]
#pragma unroll
      for (int v = 0; v < 8; v++) rf[((moff + v) * 16 + r) * 16 + n] = d[v] + bn;
      float ghr = gH[b * 16 + r];
#pragma unroll
      for (int v = 0; v < 8; v++) wacc[v] += rf[(b * 16 + r) * 16 + i0 + v] * ghr;
    }
  }
#pragma unroll
  for (int v = 0; v < 8; v++) wbf[b * 16 + i0 + v] = wacc[v];
#pragma unroll
  for (int v = 0; v < 8; v++) {
    float s = 0.f;
#pragma unroll
    for (int k = 0; k < 16; k++)
      s += rf[(b * 16 + i0 + v) * 16 + k] * wbf[b * 16 + k];
    dzr[v] -= s;
  }

  // ---- B MLP (h1 -> tb, h2 -> vb, out -> bo as Bmat flat [b][d*8+m]) ----
#pragma unroll 2
  for (int nt = 0; nt < 4; nt++) {
    v8f d = gemm_tile<16, true>(zt, 16, sw + OW1B, 16, lane, nt * 16);
    st_tanh_bias(tb, 64, lane, d, sw + OB1B, nt * 16);
  }
#pragma unroll 2
  for (int nt = 0; nt < 4; nt++) {
    v8f d = gemm_tile<64, true>(tb, 64, sw + OW2B, 64, lane, nt * 16);
    st_tanh_bias(vb, 64, lane, d, sw + OB2B, nt * 16);
  }
#pragma unroll 2
  for (int nt = 0; nt < 8; nt++) {
    v8f d = gemm_tile<64, true>(vb, 64, w3b, 64, lane, nt * 16);
    float bn = sw[OB3B + nt * 16 + n];
#pragma unroll
    for (int v = 0; v < 8; v++) bo[(moff + v) * 128 + nt * 16 + n] = d[v] + bn;
  }
#pragma unroll
  for (int v = 0; v < 8; v++) {
    float s = 0.f;
#pragma unroll
    for (int m = 0; m < 8; m++)
      s += bo[b * 128 + (i0 + v) * 8 + m] * ub[b * 8 + m];
    dzr[v] += s - EPS_C * gH[b * 16 + i0 + v];
  }

  v8f out;
#pragma unroll
  for (int v = 0; v < 8; v++) out[v] = dzr[v];
  return out;
}

// ---------------- cooperative copy fallback ----------------
__device__ __forceinline__ void cpw(float* dst, const float* src, int cnt, int tid, int nt) {
  for (int i = tid; i < cnt; i += nt) dst[i] = src[i];
}

// ---------------- kernel ----------------
__global__ __launch_bounds__(128)
void ph_heun_kernel(const float* __restrict__ z, const float* __restrict__ u,
                    const float* __restrict__ Wh1, const float* __restrict__ bh1,
                    const float* __restrict__ Wh2,
                    const float* __restrict__ W1jr, const float* __restrict__ b1jr,
                    const float* __restrict__ W2jr, const float* __restrict__ b2jr,
                    const float* __restrict__ W3jr, const float* __restrict__ b3jr,
                    const float* __restrict__ W1b, const float* __restrict__ b1b,
                    const float* __restrict__ W2b, const float* __restrict__ b2b,
                    const float* __restrict__ W3b, const float* __restrict__ b3b,
                    float* __restrict__ out, int B) {
  extern __shared__ float smem[];
  const int tid = threadIdx.x, nthr = blockDim.x;

#if HAVE_TDM
  // Tensor Data Mover: one 1D descriptor per weight array, issued by wave 0,
  // tracked with TENSORcnt; frees VALU/VMEM pipes during block warm-up.
  if (tid < 32) {
    tdm_copy_to_lds(Wh1, OWH1 * 4u, 2048);
    tdm_copy_to_lds(Wh2, OWH2 * 4u, 2048);
    tdm_copy_to_lds(bh1, OBH1 * 4u, 128);
    tdm_copy_to_lds(W1jr, OW1JR * 4u, 1024);
    tdm_copy_to_lds(b1jr, OB1JR * 4u, 64);
    tdm_copy_to_lds(W2jr, OW2JR * 4u, 4096);
    tdm_copy_to_lds(b2jr, OB2JR * 4u, 64);
    tdm_copy_to_lds(b3jr, OB3JR * 4u, 512);
    tdm_copy_to_lds(W1b, OW1B * 4u, 1024);
    tdm_copy_to_lds(b1b, OB1B * 4u, 64);
    tdm_copy_to_lds(W2b, OW2B * 4u, 4096);
    tdm_copy_to_lds(b2b, OB2B * 4u, 64);
    tdm_copy_to_lds(b3b, OB3B * 4u, 128);
  }
#else
  cpw(smem + OWH1, Wh1, 2048, tid, nthr);
  cpw(smem + OWH2, Wh2, 2048, tid, nthr);
  cpw(smem + OBH1, bh1, 128, tid, nthr);
  cpw(smem + OW1JR, W1jr, 1024, tid, nthr);
  cpw(smem + OB1JR, b1jr, 64, tid, nthr);
  cpw(smem + OW2JR, W2jr, 4096, tid, nthr);
  cpw(smem + OB2JR, b2jr, 64, tid, nthr);
  cpw(smem + OB3JR, b3jr, 512, tid, nthr);
  cpw(smem + OW1B, W1b, 1024, tid, nthr);
  cpw(smem + OB1B, b1b, 64, tid, nthr);
  cpw(smem + OW2B, W2b, 4096, tid, nthr);
  cpw(smem + OB2B, b2b, 64, tid, nthr);
  cpw(smem + OB3B, b3b, 128, tid, nthr);
#endif
  // warm L2/WGP$ for the big layer-3 weights kept in global (overlaps TDM)
  for (int i = tid; i < 512; i += nthr) __builtin_prefetch(W3jr + i * 64, 0, 0);
  for (int i = tid; i < 128; i += nthr) __builtin_prefetch(W3b + i * 64, 0, 0);
#if HAVE_TDM
  if (tid < 32) __builtin_amdgcn_s_wait_tensorcnt(0);
#endif
  __syncthreads();

  const int wave = tid >> 5, lane = tid & 31;
  float* wl = smem + WTOT + wave * WAVE_FLOATS;
  const long long rowBase = (long long)blockIdx.x * 64 + wave * 16;  // 16 rows/wave

  const float* zg = z + rowBase * 16;
  for (int i = lane; i < 256; i += 32) wl[ZT + i] = zg[i];
  const float* ug = u + rowBase * 8;
  for (int i = lane; i < 128; i += 32) wl[UB + i] = ug[i];

  const int b = lane & 15, i0 = (lane >> 4) * 8;
  float zr[8];
#pragma unroll
  for (int v = 0; v < 8; v++) zr[v] = wl[ZT + b * 16 + i0 + v];

  // Heun stage 1
  v8f dz1 = eval_step(smem, wl, W3jr, W3b, lane);
#pragma unroll
  for (int v = 0; v < 8; v++) wl[ZT + b * 16 + i0 + v] = zr[v] + dz1[v];
  // Heun stage 2 (leaves gH(z1), Bmat(z1) in LDS)
  v8f dz2 = eval_step(smem, wl, W3jr, W3b, lane);

  float* oz = out + rowBase * 16;
#pragma unroll
  for (int v = 0; v < 8; v++)
    oz[b * 16 + i0 + v] = zr[v] + 0.5f * (dz1[v] + dz2[v]);

  float* oy = out + (long long)B * 16 + rowBase * 8;
  if (lane < 16) {
#pragma unroll
    for (int m = 0; m < 8; m++) {
      float s = 0.f;
#pragma unroll
      for (int d = 0; d < 16; d++)
        s += wl[BO + b * 128 + d * 8 + m] * wl[GH + b * 16 + d];
      oy[b * 8 + m] = s;
    }
  }
}

// ---------------- host entry ----------------
extern "C" void kernel_launch(void* const* d_in, const int* in_sizes, int n_in,
                              void* d_out, int out_size, void* d_ws, size_t ws_size,
                              hipStream_t stream) {
  (void)n_in; (void)d_ws; (void)ws_size; (void)out_size;
  const float* z    = (const float*)d_in[0];
  const float* u    = (const float*)d_in[1];
  const float* Wh1  = (const float*)d_in[2];
  const float* bh1  = (const float*)d_in[3];
  const float* Wh2  = (const float*)d_in[4];
  const float* W1jr = (const float*)d_in[5];
  const float* b1jr = (const float*)d_in[6];
  const float* W2jr = (const float*)d_in[7];
  const float* b2jr = (const float*)d_in[8];
  const float* W3jr = (const float*)d_in[9];
  const float* b3jr = (const float*)d_in[10];
  const float* W1b  = (const float*)d_in[11];
  const float* b1b  = (const float*)d_in[12];
  const float* W2b  = (const float*)d_in[13];
  const float* b2b  = (const float*)d_in[14];
  const float* W3b  = (const float*)d_in[15];
  const float* b3b  = (const float*)d_in[16];
  float* out = (float*)d_out;

  const int B = in_sizes[0] / 16;        // 131072
  const int blocks = B / 64;             // 64 rows per 128-thread block (4 waves)
  const size_t shmem = (size_t)(WTOT + 4 * WAVE_FLOATS) * sizeof(float);  // ~242 KB

  (void)hipFuncSetAttribute((const void*)ph_heun_kernel,
                            hipFuncAttributeMaxDynamicSharedMemorySize,
                            (int)shmem);
  ph_heun_kernel<<<blocks, 128, shmem, stream>>>(
      z, u, Wh1, bh1, Wh2, W1jr, b1jr, W2jr, b2jr, W3jr, b3jr,
      W1b, b1b, W2b, b2b, W3b, b3b, out, B);
}